// STDPSynapse_60241211294072
// MI455X (gfx1250) — compile-verified
//
#include <hip/hip_runtime.h>

// Dimensions / constants from the reference
#define B_DIM 8
#define I_DIM 4096
#define O_DIM 4096
#define D_MAX 10
#define CHUNK 256            // K-chunk staged in LDS per iteration
#define PAD   260            // padded LDS row stride (floats): bank-conflict-free
#define A_PLUS  0.01f
#define A_MINUS 0.01f
#define ALPHA_TR 0.9512294245007140f   // exp(-1/20)

typedef float v2f __attribute__((ext_vector_type(2)));
typedef float v8f __attribute__((ext_vector_type(8)));

// ---------------------------------------------------------------------------
// Kernel 1: STDP trace update  x_new = a*x + last_pre ; y_new = a*y + last_post
// ---------------------------------------------------------------------------
__global__ void traces_k(const float* __restrict__ pre,
                         const float* __restrict__ post,
                         const float* __restrict__ xt,
                         const float* __restrict__ yt,
                         float* __restrict__ xw, float* __restrict__ yw,
                         float* __restrict__ xo, float* __restrict__ yo) {
  int i = blockIdx.x * blockDim.x + threadIdx.x;
  if (i < I_DIM) {
    float x = ALPHA_TR * xt[i] + pre[(size_t)(B_DIM - 1) * I_DIM + i];
    xw[i] = x;
    xo[i] = x;
  }
  if (i < O_DIM) {
    float y = ALPHA_TR * yt[i] + post[(size_t)(B_DIM - 1) * O_DIM + i];
    yw[i] = y;
    yo[i] = y;
  }
}

// ---------------------------------------------------------------------------
// Kernel 2: partial column sums of delay[O,I], b128 streaming
// grid: (I/1024, 16 row-splits) x 256 threads; thread owns 4 columns
// ---------------------------------------------------------------------------
__global__ void delay_partial_k(const float* __restrict__ delay,
                                float* __restrict__ part) {
  int col = blockIdx.x * 1024 + threadIdx.x * 4;
  int r0  = blockIdx.y * 256;
  float4 s = make_float4(0.f, 0.f, 0.f, 0.f);
#pragma unroll 4
  for (int r = 0; r < 256; ++r) {
    float4 d = *(const float4*)(delay + (size_t)(r0 + r) * I_DIM + col);
    s.x += d.x; s.y += d.y; s.z += d.z; s.w += d.w;
  }
  *(float4*)(part + (size_t)blockIdx.y * I_DIM + col) = s;
}

// ---------------------------------------------------------------------------
// Kernel 3: finalize mean, round-half-even, gather delayed spikes
// ---------------------------------------------------------------------------
__global__ void delay_final_k(const float* __restrict__ part,
                              const float* __restrict__ sbuf,
                              const float* __restrict__ pre,
                              float* __restrict__ outD) {
  int col = (blockIdx.x * 256 + threadIdx.x) * 4;
  float4 s = make_float4(0.f, 0.f, 0.f, 0.f);
#pragma unroll
  for (int w = 0; w < 16; ++w) {
    float4 p = *(const float4*)(part + (size_t)w * I_DIM + col);
    s.x += p.x; s.y += p.y; s.z += p.z; s.w += p.w;
  }
  const float inv = 1.0f / (float)O_DIM;
  float sv[4] = {s.x, s.y, s.z, s.w};
  float out[4];
#pragma unroll
  for (int j = 0; j < 4; ++j) {
    int idx = (int)rintf(sv[j] * inv) - 1;   // rintf = round-half-even (jnp.round)
    idx = idx < 0 ? 0 : (idx > D_MAX - 1 ? D_MAX - 1 : idx);
    // new_buffer[d] = spike_buffer[d+1] for d<9 ; new_buffer[9] = last_pre
    float v = (idx == D_MAX - 1) ? pre[(size_t)(B_DIM - 1) * I_DIM + col + j]
                                 : sbuf[(size_t)(idx + 1) * I_DIM + col + j];
    out[j] = fminf(fmaxf(v, 0.0f), 1.0f);
  }
  *(float4*)(outD + col) = make_float4(out[0], out[1], out[2], out[3]);
}

// ---------------------------------------------------------------------------
// Kernel 4 (fused, memory-bound main pass):
//   - stream weight tile [16 x CHUNK] coalesced, compute w_new, store,
//     stage raw W in LDS (padded stride, bank-conflict-free)
//   - pre-spikes tile via GLOBAL_LOAD_ASYNC_TO_LDS_B128 (ASYNCcnt path)
//   - 8 waves accumulate current = pre @ W^T with v_wmma_f32_16x16x4_f32
//   - cross-wave reduction of the 16x16 accumulator through LDS
// ---------------------------------------------------------------------------
__global__ void __launch_bounds__(256) fused_gemm_wnew_k(
    const float* __restrict__ pre,   // [B, I]
    const float* __restrict__ post,  // [B, O]
    const float* __restrict__ W,     // [O, I]
    const float* __restrict__ xw,    // x_new [I]
    const float* __restrict__ yw,    // y_new [O]
    float* __restrict__ outCur,      // [B, O]
    float* __restrict__ outW) {      // [O, I]
  __shared__ float wLDS[16 * PAD];   // raw weight tile (row = O within tile)
  __shared__ float pLDS[16 * PAD];   // pre-spikes padded to 16 rows (row = batch)
  __shared__ float red[8 * 256];     // per-wave accumulator reduction

  const int tid  = threadIdx.x;
  const int lane = tid & 31;
  const int wave = tid >> 5;
  const int o0   = blockIdx.x * 16;
  const float* lastPre = pre + (size_t)(B_DIM - 1) * I_DIM;

  // Zero the batch-padding rows (8..15) of pLDS once; never overwritten after.
  for (int idx = tid; idx < 8 * 256; idx += 256) {
    int row = 8 + (idx >> 8);
    int col = idx & 255;
    pLDS[row * PAD + col] = 0.0f;
  }

  v8f acc = {};  // 16x16 f32 accumulator fragment (M=batch-padded, N=O tile)

  for (int c0 = 0; c0 < I_DIM; c0 += CHUNK) {
    // ---- Phase 1a: async-copy pre tile rows 0..7 into LDS (2 x b128 / thread)
#pragma unroll
    for (int j = 0; j < 2; ++j) {
      int slot = tid + j * 256;            // 512 slots of 16B (8 rows x 64)
      int row  = slot >> 6;                // 0..7
      int c16  = (slot & 63) << 2;         // float column within chunk
      unsigned lds_addr = (unsigned)(size_t)(&pLDS[row * PAD + c16]);
      const float* gsrc = pre + (size_t)row * I_DIM + (size_t)(c0 + c16);
      asm volatile("global_load_async_to_lds_b128 %0, %1, off"
                   :: "v"(lds_addr), "v"(gsrc) : "memory");
    }

    // ---- Phase 1b: coalesced W tile load + w_new + LDS stage (4 float4/thread)
#pragma unroll
    for (int it = 0; it < 4; ++it) {
      int f   = tid + it * 256;        // 0..1023 float4 slots
      int row = f >> 6;                // 16 rows, 64 float4 per row
      int c4  = (f & 63) << 2;         // column (floats) within chunk
      size_t goff = (size_t)(o0 + row) * I_DIM + (size_t)(c0 + c4);
      float4 w  = *(const float4*)(W + goff);
      if (c0 + CHUNK < I_DIM) __builtin_prefetch(W + goff + CHUNK, 0, 1);
      float  sp = A_PLUS  * post[(size_t)(B_DIM - 1) * O_DIM + o0 + row];
      float  sy = A_MINUS * yw[o0 + row];
      float4 xn = *(const float4*)(xw + c0 + c4);
      float4 lp = *(const float4*)(lastPre + c0 + c4);
      float4 wn;
      wn.x = fminf(fmaxf(w.x + sp * xn.x - sy * lp.x, 0.0f), 1.0f);
      wn.y = fminf(fmaxf(w.y + sp * xn.y - sy * lp.y, 0.0f), 1.0f);
      wn.z = fminf(fmaxf(w.z + sp * xn.z - sy * lp.z, 0.0f), 1.0f);
      wn.w = fminf(fmaxf(w.w + sp * xn.w - sy * lp.w, 0.0f), 1.0f);
      *(float4*)(outW + goff) = wn;
      *(float4*)(&wLDS[row * PAD + c4]) = w;   // GEMM uses ORIGINAL weight
    }

    // Wait for this wave's async LDS fills, then rendezvous all waves.
    asm volatile("s_wait_asynccnt 0x0" ::: "memory");
    __syncthreads();

    // ---- Phase 2: each wave covers a 32-wide K slice with 8 WMMA(K=4) steps
    {
      const int m    = lane & 15;      // A fragment row (batch, padded)
      const int n    = lane & 15;      // B fragment col (O within tile)
      const int half = lane >> 4;      // K-subslot selector (K+0/1 vs K+2/3)
      const int kbase = wave * 32;
#pragma unroll
      for (int s = 0; s < 8; ++s) {
        int kk = kbase + 4 * s + 2 * half;
        // A 16x4 f32: VGPR0/1 = K{0,1} (lanes 0-15), K{2,3} (lanes 16-31)
        v2f a = *(const v2f*)(&pLDS[m * PAD + kk]);
        // B 4x16 f32: VGPR0/1 = rows K{0,1} / K{2,3}; B[k][n] = W[n][k]
        v2f b = *(const v2f*)(&wLDS[n * PAD + kk]);
        acc = __builtin_amdgcn_wmma_f32_16x16x4_f32(
            /*neg_a=*/false, a, /*neg_b=*/false, b,
            /*c_mod=*/(short)0, acc, /*reuse_a=*/false, /*reuse_b=*/false);
      }
    }
    __syncthreads();
  }

  // ---- Cross-wave reduction: 8 partial 16x16 accumulators -> output
#pragma unroll
  for (int r = 0; r < 8; ++r) red[wave * 256 + r * 32 + lane] = acc[r];
  __syncthreads();
  {
    int t = tid;
    float s = 0.0f;
#pragma unroll
    for (int w = 0; w < 8; ++w) s += red[w * 256 + t];
    int r = t >> 5;
    int l = t & 31;
    int m = r + ((l >> 4) << 3);  // VGPR r: lanes 0-15 -> M=r, lanes 16-31 -> M=r+8
    int n = l & 15;
    if (m < B_DIM) outCur[(size_t)m * O_DIM + o0 + n] = s;
  }
}

// ---------------------------------------------------------------------------
extern "C" void kernel_launch(void* const* d_in, const int* in_sizes, int n_in,
                              void* d_out, int out_size, void* d_ws, size_t ws_size,
                              hipStream_t stream) {
  (void)in_sizes; (void)n_in; (void)out_size; (void)ws_size;
  const float* pre    = (const float*)d_in[0];  // [B, I]
  const float* post   = (const float*)d_in[1];  // [B, O]
  const float* weight = (const float*)d_in[2];  // [O, I]
  const float* delay  = (const float*)d_in[3];  // [O, I]
  const float* sbuf   = (const float*)d_in[4];  // [D_MAX, I]
  const float* xt     = (const float*)d_in[5];  // [I]
  const float* yt     = (const float*)d_in[6];  // [O]

  float* out = (float*)d_out;
  // Output layout: current[B*O] | w_new[O*I] | x_new[I] | y_new[O] | delayed[I]
  float* outCur = out;
  float* outW   = out + (size_t)B_DIM * O_DIM;
  float* outX   = outW + (size_t)O_DIM * I_DIM;
  float* outY   = outX + I_DIM;
  float* outD   = outY + O_DIM;

  float* ws   = (float*)d_ws;
  float* xw   = ws;                 // x_new cache [I]
  float* yw   = ws + I_DIM;         // y_new cache [O]
  float* part = ws + I_DIM + O_DIM; // delay partial sums [16, I]

  traces_k<<<dim3(I_DIM / 256), dim3(256), 0, stream>>>(
      pre, post, xt, yt, xw, yw, outX, outY);

  delay_partial_k<<<dim3(I_DIM / 1024, O_DIM / 256), dim3(256), 0, stream>>>(
      delay, part);
  delay_final_k<<<dim3(I_DIM / 1024), dim3(256), 0, stream>>>(
      part, sbuf, pre, outD);

  fused_gemm_wnew_k<<<dim3(O_DIM / 16), dim3(256), 0, stream>>>(
      pre, post, weight, xw, yw, outCur, outW);
}